// Top1Router_90701119357093
// MI455X (gfx1250) — compile-verified
//
#include <hip/hip_runtime.h>
#include <hip/hip_bf16.h>
#include <stdint.h>

// Problem constants (static per reference): s=16384, e=8, capacity=2560
#define S_TOKENS 16384
#define NEXP     8
#define CAP      2560
#define EC       (NEXP * CAP)   // 20480 floats per token row
#define EC4      (EC / 4)       // 5120 float4 per token row per section
#define TPB      256
#define NBLK1    (S_TOKENS / TPB) // 64 routing blocks

// 16-byte int vector + address-space-qualified flavors for the async builtin:
typedef int v4i __attribute__((vector_size(16)));
typedef __attribute__((address_space(1))) v4i glb_v4i;   // global (AS1)
typedef __attribute__((address_space(3))) v4i lds_v4i;   // LDS (AS3)

#if __has_builtin(__builtin_amdgcn_ballot_w32)
#define BALLOT32(p) __builtin_amdgcn_ballot_w32(p)
#else
#define BALLOT32(p) ((unsigned)__ballot(p))
#endif

// ---------------------------------------------------------------------------
// Kernel 1: per-token top1 + softmax weight + order-preserving in-block rank
// (wave32 ballot) + per-block per-expert histogram.
// ---------------------------------------------------------------------------
__global__ __launch_bounds__(TPB) void k_route(const float* __restrict__ in,
                                               uint2* __restrict__ tokInfo,
                                               unsigned* __restrict__ blockHist) {
    const int tid = threadIdx.x;
    const int t   = blockIdx.x * TPB + tid;

    const float4* in4 = (const float4*)in;
    float4 a = in4[(size_t)t * 2 + 0];
    float4 b = in4[(size_t)t * 2 + 1];
    float v[8] = {a.x, a.y, a.z, a.w, b.x, b.y, b.z, b.w};

    // argmax (first occurrence, like jnp.argmax) and softmax value at top1:
    int ex = 0; float m = v[0];
    #pragma unroll
    for (int i = 1; i < 8; ++i) { if (v[i] > m) { m = v[i]; ex = i; } }
    float sum = 0.f;
    #pragma unroll
    for (int i = 0; i < 8; ++i) sum += expf(v[i] - m);
    float w = 1.0f / sum;   // exp(max-max)/sum

    // wave32 order-preserving rank within wave, per-expert counts
    const unsigned lane = tid & 31u;
    const unsigned wv   = tid >> 5;
    const unsigned ltmask = (1u << lane) - 1u;
    unsigned myrank = 0, cnt8 = 0;
    #pragma unroll
    for (int q = 0; q < 8; ++q) {
        bool p = (ex == q);
        unsigned bal = BALLOT32(p);
        if (p) myrank = __popc(bal & ltmask);
        if (lane == (unsigned)q) cnt8 = __popc(bal);  // lane q holds count of expert q
    }

    __shared__ unsigned wcnt[8][8];  // [wave][expert]
    if (lane < 8) wcnt[wv][lane] = cnt8;
    __syncthreads();

    unsigned before = 0;
    for (unsigned u = 0; u < wv; ++u) before += wcnt[u][ex];
    const unsigned rblk = before + myrank;   // rank within this 256-token block

    tokInfo[t] = make_uint2(__float_as_uint(w), (unsigned)ex | (rblk << 8));

    if (tid < 8) {
        unsigned tot = 0;
        #pragma unroll
        for (int u = 0; u < 8; ++u) tot += wcnt[u][tid];
        blockHist[blockIdx.x * NEXP + tid] = tot;
    }
}

// ---------------------------------------------------------------------------
// Kernel 2: exclusive prefix over blocks, per expert (64 x 8, trivial).
// ---------------------------------------------------------------------------
__global__ void k_scan(const unsigned* __restrict__ blockHist,
                       unsigned* __restrict__ blockOff) {
    const int q = threadIdx.x;
    if (q < NEXP) {
        unsigned run = 0;
        for (int b = 0; b < NBLK1; ++b) {
            blockOff[b * NEXP + q] = run;
            run += blockHist[b * NEXP + q];
        }
    }
}

// ---------------------------------------------------------------------------
// Kernel 3: stream the dense output. One workgroup per token writes its
// [8,2560] weights row and [8,2560] mask row: zeros via async LDS->global
// B128 stores (CDNA5 async engine, ASYNCcnt), hot element via plain store
// to a disjoint address.
// ---------------------------------------------------------------------------
__global__ __launch_bounds__(TPB) void k_fill(const uint2* __restrict__ tokInfo,
                                              const unsigned* __restrict__ blockOff,
                                              float* __restrict__ out) {
    const int t   = blockIdx.x;
    const int tid = threadIdx.x;

    const uint2 info = tokInfo[t];
    const float    w    = __uint_as_float(info.x);
    const unsigned ex   = info.y & 7u;
    const unsigned rblk = info.y >> 8;
    const unsigned rank = blockOff[(t >> 8) * NEXP + ex] + rblk;
    const bool     valid = rank < (unsigned)CAP;
    const unsigned hot = ex * CAP + rank;          // flat index in [e*c)
    const unsigned hv  = hot >> 2;                 // float4 index
    const unsigned comp = hot & 3u;

    float4* wb = (float4*)out + (size_t)t * EC4;
    float4* mb = (float4*)out + (size_t)S_TOKENS * EC4 + (size_t)t * EC4;

    __shared__ __align__(16) float zbuf[4];
    if (tid < 4) zbuf[tid] = 0.0f;
    __syncthreads();

#if __has_builtin(__builtin_amdgcn_global_store_async_from_lds_b128)
    lds_v4i* ldsz = (lds_v4i*)(&zbuf[0]);
    for (int j = tid; j < EC4; j += TPB) {
        const bool isHot = valid && ((unsigned)j == hv);
        if (!isHot) {
            __builtin_amdgcn_global_store_async_from_lds_b128(
                (glb_v4i*)(wb + j), ldsz, 0, 0);
            __builtin_amdgcn_global_store_async_from_lds_b128(
                (glb_v4i*)(mb + j), ldsz, 0, 0);
        }
    }
#else
    // Inline-asm fallback (saddr form: voffset, vdata(lds addr), s[base])
    const unsigned ldsoff = (unsigned)(uintptr_t)(
        (__attribute__((address_space(3))) float*)(&zbuf[0]));
    const unsigned long long wbase = (unsigned long long)wb;
    const unsigned long long mbase = (unsigned long long)mb;
    for (int j = tid; j < EC4; j += TPB) {
        const bool isHot = valid && ((unsigned)j == hv);
        if (!isHot) {
            const unsigned off = (unsigned)j * 16u;
            asm volatile("global_store_async_from_lds_b128 %0, %1, %2"
                         :: "v"(off), "v"(ldsoff), "s"(wbase) : "memory");
            asm volatile("global_store_async_from_lds_b128 %0, %1, %2"
                         :: "v"(off), "v"(ldsoff), "s"(mbase) : "memory");
        }
    }
#endif

    // Hot elements (addresses disjoint from the async-stored float4s above).
    if (tid == 0 && valid) {
        float4 zw = make_float4(0.f, 0.f, 0.f, 0.f);
        ((float*)&zw)[comp] = w;
        wb[hv] = zw;
        float4 zm = make_float4(0.f, 0.f, 0.f, 0.f);
        ((float*)&zm)[comp] = 1.0f;
        mb[hv] = zm;
    }

    // Drain the async engine before wave end (hygiene; S_ENDPGM also waits).
#if __has_builtin(__builtin_amdgcn_s_wait_asynccnt)
    __builtin_amdgcn_s_wait_asynccnt(0);
#else
    asm volatile("s_wait_asynccnt 0" ::: "memory");
#endif
}

// ---------------------------------------------------------------------------
extern "C" void kernel_launch(void* const* d_in, const int* in_sizes, int n_in,
                              void* d_out, int out_size, void* d_ws, size_t ws_size,
                              hipStream_t stream) {
    const float* in  = (const float*)d_in[0];
    float*       out = (float*)d_out;

    // workspace layout: tokInfo (16384*8B) | blockHist (64*8*4B) | blockOff
    uint2*    tokInfo   = (uint2*)d_ws;
    unsigned* blockHist = (unsigned*)((char*)d_ws + (size_t)S_TOKENS * 8);
    unsigned* blockOff  = blockHist + NBLK1 * NEXP;

    k_route<<<NBLK1, TPB, 0, stream>>>(in, tokInfo, blockHist);
    k_scan<<<1, 32, 0, stream>>>(blockHist, blockOff);
    k_fill<<<S_TOKENS, TPB, 0, stream>>>(tokInfo, blockOff, out);
}